// GCN_GRU_3959959847414
// MI455X (gfx1250) — compile-verified
//
#include <hip/hip_runtime.h>
#include <hip/hip_bf16.h>

// ---------------------------------------------------------------------------
// GCN (self-loop, sym-norm) -> 1-step GRU (h0=0) -> linear head, fp32 on
// gfx1250. Dense GEMMs use V_WMMA_F32_16X16X4_F32 (exact fp32 matrix path).
// Edge scatter uses coalesced gathers + L2-resident f32 atomics.
// ---------------------------------------------------------------------------

typedef float v2f __attribute__((ext_vector_type(2)));
typedef float v8f __attribute__((ext_vector_type(8)));

#define GG_N 100000
#define GG_E 3200000
#define GG_T 12
#define GG_H 64

__device__ __forceinline__ v8f gg_wmma_f32(v2f a, v2f b, v8f c) {
    // D = A(16x4 f32) x B(4x16 f32) + C(16x16 f32)
    return __builtin_amdgcn_wmma_f32_16x16x4_f32(
        /*neg_a=*/false, a, /*neg_b=*/false, b,
        /*c_mod=*/(short)0, c, /*reuse_a=*/false, /*reuse_b=*/false);
}

// -------------------- K1: zero g, deg = 1.0 (self-loop) --------------------
__global__ void gg_init_kernel(float* __restrict__ g, float* __restrict__ deg) {
    int i = blockIdx.x * blockDim.x + threadIdx.x;      // over N*H
    if (i >= GG_N * GG_H) return;
    g[i] = 0.0f;
    if ((i & (GG_H - 1)) == 0) deg[i >> 6] = 1.0f;
}

// -------------------- K2: in-degree accumulation ---------------------------
__global__ void gg_deg_kernel(const int* __restrict__ ei, float* __restrict__ deg) {
    int e = blockIdx.x * blockDim.x + threadIdx.x;
    if (e >= GG_E) return;
    int d = ei[GG_E + e];                               // edge_idx[1][e]
    atomicAdd(&deg[d], 1.0f);
}

// -------------------- K3: dinv = rsqrt(deg) --------------------------------
__global__ void gg_dinv_kernel(const float* __restrict__ deg, float* __restrict__ dinv) {
    int i = blockIdx.x * blockDim.x + threadIdx.x;
    if (i >= GG_N) return;
    float dg = deg[i];
    dinv[i] = dg > 0.0f ? __frsqrt_rn(dg) : 0.0f;
}

// -------------------- K4: xw = x_seq @ W_gcn via fp32 WMMA -----------------
// One wave per 16x16 output tile; K=12 accumulated as 3 WMMA(K=4) steps.
__global__ void gg_xw_wmma_kernel(const float* __restrict__ x,
                                  const float* __restrict__ Wg,
                                  float* __restrict__ xw, int nTiles) {
    int wave = (blockIdx.x * blockDim.x + threadIdx.x) >> 5;
    if (wave >= nTiles) return;                          // uniform per wave
    int lane = threadIdx.x & 31;
    int col  = lane & 15;                                // M for A, N for B/C/D
    int hi   = lane >> 4;
    int kb   = hi * 2;                                   // K sub-pair within step
    int m0   = (wave >> 2) * 16;
    int n0   = (wave & 3) * 16;

    v8f c = {};
#pragma unroll
    for (int s = 0; s < 3; ++s) {
        int k0 = s * 4 + kb;                             // 0..11
        v2f a, b;
        a.x = x[(m0 + col) * GG_T + k0];
        a.y = x[(m0 + col) * GG_T + k0 + 1];
        b.x = Wg[(k0    ) * GG_H + n0 + col];
        b.y = Wg[(k0 + 1) * GG_H + n0 + col];
        c = gg_wmma_f32(a, b, c);
    }
#pragma unroll
    for (int r = 0; r < 8; ++r)
        xw[(size_t)(m0 + r + hi * 8) * GG_H + n0 + col] = c[r];
}

// -------------------- K5: edge scatter (dominant traffic) ------------------
// 64 consecutive lanes per edge: coalesced 256B gather of xw[src], 64 f32
// atomics into g[dst] (resolved in L2; g is 25.6MB << 192MB L2).
__global__ void gg_edge_scatter_kernel(const int* __restrict__ ei,
                                       const float* __restrict__ xw,
                                       const float* __restrict__ dinv,
                                       float* __restrict__ g) {
    int e = blockIdx.x * 4 + (threadIdx.x >> 6);         // 4 edges / 256-thr block
    int h = threadIdx.x & 63;
    if (e >= GG_E) return;
    int s = ei[e];                                       // src
    int d = ei[GG_E + e];                                // dst
    float nr = dinv[s] * dinv[d];
    atomicAdd(&g[(size_t)d * GG_H + h], xw[(size_t)s * GG_H + h] * nr);
}

// -------------------- K6: self-loop message + GCN bias ---------------------
__global__ void gg_selfbias_kernel(const float* __restrict__ xw,
                                   const float* __restrict__ dinv,
                                   const float* __restrict__ b_gcn,
                                   float* __restrict__ g) {
    int i = blockIdx.x * blockDim.x + threadIdx.x;       // over N*H
    if (i >= GG_N * GG_H) return;
    int node = i >> 6;
    int h    = i & 63;
    float di = dinv[node];
    g[i] += xw[i] * di * di + b_gcn[h];
}

// -------------------- K7: fused GRU(h0=0) + linear head via fp32 WMMA ------
// One wave per 16 nodes. A-fragments (g tile, 16x64) preloaded once; 4 column
// tiles x 3 gates x 16 K-steps = 192 WMMAs; gates fused in-register; head dot
// product reduced with wave32 xor-shuffles.
__global__ void gg_gru_head_kernel(const float* __restrict__ g,
                                   const float* __restrict__ w_ih,
                                   const float* __restrict__ b_ih,
                                   const float* __restrict__ b_hh,
                                   const float* __restrict__ W_fc,
                                   const float* __restrict__ b_fc,
                                   float* __restrict__ out, int nTiles) {
    int wave = (blockIdx.x * blockDim.x + threadIdx.x) >> 5;
    if (wave >= nTiles) return;                          // uniform per wave
    int lane = threadIdx.x & 31;
    int col  = lane & 15;
    int hi   = lane >> 4;
    int kb   = hi * 2;
    size_t m0 = (size_t)wave * 16;

    // Preload A: 16 K-steps of the 16x64 g tile (32 VGPRs).
    v2f A[16];
#pragma unroll
    for (int s = 0; s < 16; ++s) {
        const float* gp = g + (m0 + col) * GG_H + s * 4 + kb;
        A[s].x = gp[0];
        A[s].y = gp[1];
    }

    v8f acc = {};
    for (int jt = 0; jt < 4; ++jt) {
        int jr = jt * 16 + col;                          // reset-gate col (0..63)
        int jz = GG_H + jr;                              // update-gate col
        int jn = 2 * GG_H + jr;                          // new-gate col
        v8f cr = {}, cz = {}, cn = {};
#pragma unroll
        for (int s = 0; s < 16; ++s) {
            int k0 = s * 4 + kb;
            v2f br, bz, bn;
            br.x = w_ih[jr * GG_H + k0]; br.y = w_ih[jr * GG_H + k0 + 1];
            bz.x = w_ih[jz * GG_H + k0]; bz.y = w_ih[jz * GG_H + k0 + 1];
            bn.x = w_ih[jn * GG_H + k0]; bn.y = w_ih[jn * GG_H + k0 + 1];
            cr = gg_wmma_f32(A[s], br, cr);
            cz = gg_wmma_f32(A[s], bz, cz);
            cn = gg_wmma_f32(A[s], bn, cn);
        }
        float bR  = b_ih[jr] + b_hh[jr];
        float bZ  = b_ih[jz] + b_hh[jz];
        float bIn = b_ih[jn];
        float bHn = b_hh[jn];
        float wf  = W_fc[jr];                            // head weight for this col
#pragma unroll
        for (int e = 0; e < 8; ++e) {
            float rv = 1.0f / (1.0f + __expf(-(cr[e] + bR)));
            float zv = 1.0f / (1.0f + __expf(-(cz[e] + bZ)));
            float nv = tanhf(cn[e] + bIn + rv * bHn);
            acc[e] += (1.0f - zv) * nv * wf;             // h' = (1-z)*n, z*h0 = 0
        }
    }

    // Reduce head partials across each 16-lane half (xor masks keep bit4).
#pragma unroll
    for (int m = 1; m < 16; m <<= 1) {
#pragma unroll
        for (int e = 0; e < 8; ++e)
            acc[e] += __shfl_xor(acc[e], m, 32);
    }
    if (col == 0) {
        float bf = b_fc[0];
#pragma unroll
        for (int e = 0; e < 8; ++e)
            out[m0 + e + hi * 8] = acc[e] + bf;
    }
}

// ---------------------------------------------------------------------------
extern "C" void kernel_launch(void* const* d_in, const int* in_sizes, int n_in,
                              void* d_out, int out_size, void* d_ws, size_t ws_size,
                              hipStream_t stream) {
    const float* x_seq = (const float*)d_in[0];
    const int*   e_idx = (const int*)  d_in[1];
    const float* W_gcn = (const float*)d_in[2];
    const float* b_gcn = (const float*)d_in[3];
    const float* w_ih  = (const float*)d_in[4];
    // d_in[5] = w_hh: unused (h0 == 0)
    const float* b_ih  = (const float*)d_in[6];
    const float* b_hh  = (const float*)d_in[7];
    const float* W_fc  = (const float*)d_in[8];
    const float* b_fc  = (const float*)d_in[9];
    float* out = (float*)d_out;

    // Workspace layout (floats): deg | dinv | xw[N*H] | g[N*H]  (~52 MB)
    const size_t NPAD = 100352;                          // N rounded up to a multiple of 2048
    float* ws   = (float*)d_ws;
    float* deg  = ws;
    float* dinv = ws + NPAD;
    float* xw   = ws + 2 * NPAD;
    float* gbuf = xw + (size_t)GG_N * GG_H;

    const int nh_blocks = (GG_N * GG_H + 255) / 256;     // 25000

    // K1: init g = 0, deg = 1 (self-loop)
    gg_init_kernel<<<nh_blocks, 256, 0, stream>>>(gbuf, deg);
    // K2: in-degree atomics over dst
    gg_deg_kernel<<<(GG_E + 255) / 256, 256, 0, stream>>>(e_idx, deg);
    // K3: dinv = rsqrt(deg)
    gg_dinv_kernel<<<(GG_N + 255) / 256, 256, 0, stream>>>(deg, dinv);
    // K4: xw = x_seq @ W_gcn  (WMMA fp32), 6250 M-tiles x 4 N-tiles
    {
        int nTiles = (GG_N / 16) * (GG_H / 16);          // 25000 waves
        int blocks = (nTiles * 32 + 255) / 256;
        gg_xw_wmma_kernel<<<blocks, 256, 0, stream>>>(x_seq, W_gcn, xw, nTiles);
    }
    // K5: edge gather/scatter (dominant traffic; L2-resident atomics)
    gg_edge_scatter_kernel<<<GG_E / 4, 256, 0, stream>>>(e_idx, xw, dinv, gbuf);
    // K6: self-loop message + bias
    gg_selfbias_kernel<<<nh_blocks, 256, 0, stream>>>(xw, dinv, b_gcn, gbuf);
    // K7: fused GRU + head (WMMA fp32), one wave per 16 nodes
    {
        int nTiles = GG_N / 16;                          // 6250 waves
        int blocks = (nTiles * 32 + 255) / 256;
        gg_gru_head_kernel<<<blocks, 256, 0, stream>>>(gbuf, w_ih, b_ih, b_hh,
                                                       W_fc, b_fc, out, nTiles);
    }
}